// SpatialTransformer_16527034155639
// MI455X (gfx1250) — compile-verified
//
#include <hip/hip_runtime.h>

// Spatial transformer (homography warp + bilinear sample) for MI455X (gfx1250).
//
// Memory-bound problem (~0.5 GB traffic -> ~20us floor @ 23.3 TB/s HBM), but
// round-1 disasm showed IEEE-divide chains pushing VALU time to the same order
// as the HBM floor. This round: v_rcp_f32 instead of IEEE divide, and the
// normalized->pixel mapping (v+1)*64 folded into the WMMA coefficients so the
// WMMA outputs are directly pixel-space numerators.
//
// Coordinate transform per wave (one 16x16 output tile) via V_WMMA_F32_16X16X4_F32:
//   A (16x4), row m  = [ y_t(m), 1, 0, 0 ]
//   B (4x16), col n  = [ c_r1, c_r0*x_t(n) + c_r2, 0, 0 ]
//   D[m][n] = c_r1*y_t(m) + c_r0*x_t(n) + c_r2
// with c_X = 64*(H_row0 + H_row2), c_Y = 64*(H_row1 + H_row2), c_Z = H_row2,
// so fx = D_X/D_Z, fy = D_Y/D_Z are already in pixel space.
//
// VGPR layouts (CDNA5 ISA 7.12.2, wave32):
//   A 16x4 f32 : lanes 0-15 -> M=lane, vgpr0=K0, vgpr1=K1; lanes 16-31 -> K2,K3 (zeros)
//   B 4x16 f32 : lanes 0-15 -> N=lane, vgpr0=K0, vgpr1=K1; lanes 16-31 -> K2,K3 (zeros)
//   D 16x16 f32: vgpr j, lanes 0-15 -> M=j, N=lane; lanes 16-31 -> M=j+8, N=lane-16

typedef __attribute__((ext_vector_type(2))) float v2f;
typedef __attribute__((ext_vector_type(8))) float v8f;

__global__ __launch_bounds__(256) void SpatialTransformer_16527034155639_kernel(
    const float* __restrict__ image,   // [B,128,128,3]
    const float* __restrict__ Hmat,    // [B,3,3]
    float* __restrict__ out)           // [B,128,128,1]
{
    const int lane  = threadIdx.x & 31;
    const int wave  = threadIdx.x >> 5;
    const int batch = blockIdx.x >> 3;                 // 8 blocks per batch
    const int tile  = ((blockIdx.x & 7) << 3) | wave;  // 0..63 (8x8 tiles of 16x16)
    const int ty0   = (tile >> 3) << 4;
    const int tx0   = (tile & 7)  << 4;

    // H is uniform per block -> compiler scalarizes these loads.
    const float* Hb = Hmat + batch * 9;
    const float h20 = Hb[6], h21 = Hb[7], h22 = Hb[8];
    // Pixel-space numerator coefficients: fx_num = 64*(row0 + row2), fy_num = 64*(row1 + row2)
    const float cx0 = 64.0f * (Hb[0] + h20), cx1 = 64.0f * (Hb[1] + h21), cx2 = 64.0f * (Hb[2] + h22);
    const float cy0 = 64.0f * (Hb[3] + h20), cy1 = 64.0f * (Hb[4] + h21), cy2 = 64.0f * (Hb[5] + h22);

    const float step = 2.0f / 127.0f;                  // linspace(-1,1,128)
    const int   lo   = lane & 15;
    const bool  hi   = lane >= 16;

    // A: row m = [y_t(m), 1, 0, 0]
    v2f a;
    a.x = hi ? 0.0f : (-1.0f + step * (float)(ty0 + lo));
    a.y = hi ? 0.0f : 1.0f;

    const float xt = -1.0f + step * (float)(tx0 + lo);

    // B per component: col n = [c_r1, c_r0*x_t(n)+c_r2, 0, 0]
    v2f bx, by, bz;
    bx.x = hi ? 0.0f : cx1;  bx.y = hi ? 0.0f : fmaf(cx0, xt, cx2);
    by.x = hi ? 0.0f : cy1;  by.y = hi ? 0.0f : fmaf(cy0, xt, cy2);
    bz.x = hi ? 0.0f : h21;  bz.y = hi ? 0.0f : fmaf(h20, xt, h22);

    v8f czero = {};
    // 8 args: (neg_a, A, neg_b, B, c_mod, C, reuse_a, reuse_b)
    v8f X = __builtin_amdgcn_wmma_f32_16x16x4_f32(false, a, false, bx, (short)0, czero, false, false);
    v8f Y = __builtin_amdgcn_wmma_f32_16x16x4_f32(false, a, false, by, (short)0, czero, false, false);
    v8f Z = __builtin_amdgcn_wmma_f32_16x16x4_f32(false, a, false, bz, (short)0, czero, false, false);

    const int px      = tx0 + lo;
    const int py_base = ty0 + (hi ? 8 : 0);
    const float* __restrict__ imgb = image + (size_t)batch * (128 * 128 * 3);
    float* __restrict__       outb = out   + (size_t)batch * (128 * 128);

#pragma unroll
    for (int j = 0; j < 8; ++j) {
        // v_rcp_f32 (<=1 ulp) instead of IEEE divide chain: coordinate error
        // ~1e-5 px -> output error ~1e-5 abs; saves ~8 VALU per pixel.
        const float rz = __builtin_amdgcn_rcpf(Z[j]);
        const float fx = X[j] * rz;                    // already pixel space
        const float fy = Y[j] * rz;

        const float f0x = floorf(fx);
        const float f0y = floorf(fy);
        // Clamp in float (min/max pairs), convert once.
        const float x0f = fminf(fmaxf(f0x,        0.0f), 127.0f);
        const float x1f = fminf(fmaxf(f0x + 1.0f, 0.0f), 127.0f);
        const float y0f = fminf(fmaxf(f0y,        0.0f), 127.0f);
        const float y1f = fminf(fmaxf(f0y + 1.0f, 0.0f), 127.0f);
        const int x0 = (int)x0f, x1 = (int)x1f;
        const int y0 = (int)y0f, y1 = (int)y1f;

        // Weights from CLIPPED coords (matches reference exactly).
        const float wa = (x1f - fx) * (y1f - fy);
        const float wb = (x1f - fx) * (fy - y0f);
        const float wc = (fx - x0f) * (y1f - fy);
        const float wd = (fx - x0f) * (fy - y0f);

        // Channel-0 gathers (NHWC, C=3 -> stride 3 floats).
        const int r0 = y0 * 384, r1 = y1 * 384;        // 128*3
        const float Ia = imgb[r0 + x0 * 3];
        const float Ib = imgb[r1 + x0 * 3];
        const float Ic = imgb[r0 + x1 * 3];
        const float Id = imgb[r1 + x1 * 3];

        outb[(py_base + j) * 128 + px] = wa * Ia + wb * Ib + wc * Ic + wd * Id;
    }
}

extern "C" void kernel_launch(void* const* d_in, const int* in_sizes, int n_in,
                              void* d_out, int out_size, void* d_ws, size_t ws_size,
                              hipStream_t stream) {
    (void)in_sizes; (void)n_in; (void)d_ws; (void)ws_size; (void)out_size;
    const float* image = (const float*)d_in[0];   // [2048,128,128,3] f32
    const float* Hmat  = (const float*)d_in[1];   // [2048,3,3] f32
    float*       out   = (float*)d_out;           // [2048,128,128,1] f32

    const int B = 2048;
    // 8 waves/block = 8 tiles/block; 64 tiles/image -> 8 blocks/image.
    SpatialTransformer_16527034155639_kernel<<<B * 8, 256, 0, stream>>>(image, Hmat, out);
}